// GAT_65274912964682
// MI455X (gfx1250) — compile-verified
//
#include <hip/hip_runtime.h>

#define NEG_SLOPE 0.2f

typedef float v2f __attribute__((ext_vector_type(2)));
typedef float v8f __attribute__((ext_vector_type(8)));

static __host__ int cdiv_i(long long a, long long b) { return (int)((a + b - 1) / b); }

// ---- order-preserving float<->uint mapping for atomicMax on f32 ----
static __device__ __forceinline__ unsigned f2ord(float f) {
    unsigned u = __float_as_uint(f);
    return (u & 0x80000000u) ? ~u : (u | 0x80000000u);
}
static __device__ __forceinline__ float ord2f(unsigned u) {
    return (u & 0x80000000u) ? __uint_as_float(u & 0x7fffffffu) : __uint_as_float(~u);
}

// =====================================================================
// FP32 WMMA GEMM: C[rows x NCOLS] = A[rows x KDIM] * B[KDIM x NCOLS]
// One wave computes a 16 x (16*NT) strip using v_wmma_f32_16x16x4_f32.
// A-fragment (16x4 f32): lanes 0-15 row M=lane, K = {k, k+1};
//                        lanes 16-31 row M=lane-16, K = {k+2, k+3}.
// B-fragment (4x16 f32): mirrored (N = lane&15, same K split).
// C/D (16x16 f32, 8 VGPRs): lanes 0-15 -> M = i, lanes 16-31 -> M = i+8,
//                           N = lane&15.
// =====================================================================
template <int KDIM, int NCOLS, int NT>
__global__ void __launch_bounds__(256) gemm_wmma_f32(
    const float* __restrict__ A, const float* __restrict__ B,
    float* __restrict__ C, int rowTiles)
{
    constexpr int STRIPS = NCOLS / (16 * NT);
    const int wave = (int)((blockIdx.x * blockDim.x + threadIdx.x) >> 5);
    const int lane = (int)(threadIdx.x & 31);
    const int rowTile = wave / STRIPS;
    const int strip   = wave - rowTile * STRIPS;
    if (rowTile >= rowTiles) return;   // wave-uniform: EXEC stays all-ones

    const int row0 = rowTile * 16;
    const int col0 = strip * 16 * NT;
    const int mrow = lane & 15;
    const int kb   = (lane >> 4) << 1;   // 0 or 2

    v8f acc[NT];
#pragma unroll
    for (int t = 0; t < NT; ++t) acc[t] = (v8f)0.0f;

    const float* arow = A + (size_t)(row0 + mrow) * KDIM;
#pragma unroll 4
    for (int k = 0; k < KDIM; k += 4) {
        v2f a;
        a.x = arow[k + kb];
        a.y = arow[k + kb + 1];
#pragma unroll
        for (int t = 0; t < NT; ++t) {
            const float* bp = B + (size_t)(k + kb) * NCOLS + col0 + t * 16 + mrow;
            v2f b;
            b.x = bp[0];
            b.y = bp[NCOLS];
            acc[t] = __builtin_amdgcn_wmma_f32_16x16x4_f32(
                false, a, false, b, (short)0, acc[t], false, false);
        }
    }

    const int rbase = row0 + ((lane >> 4) << 3);
#pragma unroll
    for (int t = 0; t < NT; ++t) {
#pragma unroll
        for (int i = 0; i < 8; ++i) {
            C[(size_t)(rbase + i) * NCOLS + col0 + t * 16 + mrow] = acc[t][i];
        }
    }
}

// ---- per-(node,head) attention scores: a = sum_c h[n,h,c] * att[h,c] ----
template <int H, int C>
__global__ void att_scores(const float* __restrict__ h,
                           const float* __restrict__ att_s,
                           const float* __restrict__ att_d,
                           float* __restrict__ as_out,
                           float* __restrict__ ad_out, int n)
{
    int idx = blockIdx.x * blockDim.x + threadIdx.x;
    if (idx >= n * H) return;
    int node = idx / H, head = idx % H;
    const float* hp = h + (size_t)node * (H * C) + head * C;
    float ss = 0.f, sd = 0.f;
#pragma unroll
    for (int c = 0; c < C; ++c) {
        float v = hp[c];
        ss += v * att_s[head * C + c];
        sd += v * att_d[head * C + c];
    }
    as_out[idx] = ss;
    ad_out[idx] = sd;
}

// ---- self-loop-augmented edge lookup ----
static __device__ __forceinline__ void edge_st(const int* __restrict__ ei,
                                               int E, int e, int& s, int& t)
{
    if (e < E) { s = ei[e]; t = ei[E + e]; }
    else       { s = e - E; t = e - E; }
}

// ---- pass A: leaky-relu logits + segment max via ordered atomicMax ----
template <int H>
__global__ void edge_maxpass(const int* __restrict__ ei, int E, int ET,
                             const float* __restrict__ as_,
                             const float* __restrict__ ad_,
                             float* __restrict__ ebuf,
                             unsigned* __restrict__ mbuf)
{
    long long gid = blockIdx.x * (long long)blockDim.x + threadIdx.x;
    if (gid >= (long long)ET * H) return;
    int e = (int)(gid / H), h = (int)(gid % H);
    int s, t; edge_st(ei, E, e, s, t);
    float v = as_[s * H + h] + ad_[t * H + h];
    v = v > 0.f ? v : NEG_SLOPE * v;
    ebuf[gid] = v;
    atomicMax(&mbuf[t * H + h], f2ord(v));
}

// ---- pass B: exp(e - max) + segment sum of denominators ----
template <int H>
__global__ void edge_exppass(const int* __restrict__ ei, int E, int ET,
                             const unsigned* __restrict__ mbuf,
                             float* __restrict__ ebuf,
                             float* __restrict__ denom)
{
    long long gid = blockIdx.x * (long long)blockDim.x + threadIdx.x;
    if (gid >= (long long)ET * H) return;
    int e = (int)(gid / H), h = (int)(gid % H);
    int s, t; edge_st(ei, E, e, s, t);
    (void)s;
    float ex = __expf(ebuf[gid] - ord2f(mbuf[t * H + h]));
    ebuf[gid] = ex;
    atomicAdd(&denom[t * H + h], ex);
}

// ---- pass C: alpha-weighted gather of h[src] + scatter-add to agg[tgt] ----
template <int H, int C>
__global__ void edge_scatter(const int* __restrict__ ei, int E, int ET,
                             const float* __restrict__ hbuf,
                             const float* __restrict__ ebuf,
                             const float* __restrict__ denom,
                             float* __restrict__ agg)
{
    constexpr int G = (H * C) / 4;   // float4 groups per edge
    long long gid = blockIdx.x * (long long)blockDim.x + threadIdx.x;
    if (gid >= (long long)ET * G) return;
    int e = (int)(gid / G), q = (int)(gid % G);
    int h = (q * 4) / C;
    int s, t; edge_st(ei, E, e, s, t);
    float alpha = ebuf[(size_t)e * H + h] / denom[t * H + h];
    const float4 v = ((const float4*)hbuf)[(size_t)s * G + q];
    float* dst = agg + (size_t)t * (H * C) + q * 4;
    atomicAdd(dst + 0, v.x * alpha);
    atomicAdd(dst + 1, v.y * alpha);
    atomicAdd(dst + 2, v.z * alpha);
    atomicAdd(dst + 3, v.w * alpha);
}

// ---- epilogue: out = elu(in + bias) (in-place safe) ----
template <int HC>
__global__ void bias_elu(const float* __restrict__ in,
                         const float* __restrict__ b,
                         float* __restrict__ out, long long n)
{
    long long i = blockIdx.x * (long long)blockDim.x + threadIdx.x;
    if (i >= n) return;
    float v = in[i] + b[(int)(i % HC)];
    out[i] = v > 0.f ? v : (__expf(v) - 1.f);
}

__global__ void fill_f32(float* __restrict__ p, float v, long long n)
{
    long long i = blockIdx.x * (long long)blockDim.x + threadIdx.x;
    if (i < n) p[i] = v;
}

// =====================================================================
extern "C" void kernel_launch(void* const* d_in, const int* in_sizes, int n_in,
                              void* d_out, int out_size, void* d_ws, size_t ws_size,
                              hipStream_t stream)
{
    (void)n_in; (void)out_size; (void)ws_size;
    const float* x    = (const float*)d_in[0];
    const int*   ei   = (const int*)  d_in[1];
    const float* W1   = (const float*)d_in[2];
    const float* b1   = (const float*)d_in[3];
    const float* atS1 = (const float*)d_in[4];
    const float* atD1 = (const float*)d_in[5];
    const float* W2   = (const float*)d_in[6];
    const float* b2   = (const float*)d_in[7];
    const float* atS2 = (const float*)d_in[8];
    const float* atD2 = (const float*)d_in[9];

    const int FIN = 128, H1 = 8, C1 = 32, HC1 = H1 * C1;   // 256
    const int H2 = 1, C2 = 32, HC2 = H2 * C2;              // 32
    const int N  = in_sizes[0] / FIN;
    const int E  = in_sizes[1] / 2;
    const int ET = E + N;                                   // + self loops

    // ---- workspace carving (floats) ----
    float* ws   = (float*)d_ws;
    float* h1   = ws;                               // N*256
    float* aS1  = h1  + (size_t)N * HC1;            // N*8
    float* aD1  = aS1 + (size_t)N * H1;             // N*8
    float* m1   = aD1 + (size_t)N * H1;             // N*8 (ordered uints)
    float* den1 = m1  + (size_t)N * H1;             // N*8
    float* e1   = den1 + (size_t)N * H1;            // ET*8
    float* g1   = e1  + (size_t)ET * H1;            // N*256 (agg1 -> elu in place)
    float* h2   = g1  + (size_t)N * HC1;            // N*32
    float* aS2  = h2  + (size_t)N * HC2;            // N
    float* aD2  = aS2 + (size_t)N;                  // N
    float* m2   = aD2 + (size_t)N;                  // N
    float* den2 = m2  + (size_t)N;                  // N
    float* e2   = den2 + (size_t)N;                 // ET
    float* out  = (float*)d_out;                    // agg2 accumulated in place

    const int TB = 256;

    // ================= layer 1 =================
    {   // h1 = x @ W1  (rowTiles=N/16, 4 strips of 64 cols)
        int rowTiles = N / 16;
        long long waves = (long long)rowTiles * (HC1 / 64);
        gemm_wmma_f32<128, 256, 4><<<cdiv_i(waves * 32, TB), TB, 0, stream>>>(x, W1, h1, rowTiles);
    }
    att_scores<H1, C1><<<cdiv_i((long long)N * H1, TB), TB, 0, stream>>>(h1, atS1, atD1, aS1, aD1, N);

    fill_f32<<<cdiv_i((long long)N * H1 * 2, TB), TB, 0, stream>>>(m1, 0.f, (long long)N * H1 * 2); // m1+den1
    fill_f32<<<cdiv_i((long long)N * HC1, TB), TB, 0, stream>>>(g1, 0.f, (long long)N * HC1);

    edge_maxpass<H1><<<cdiv_i((long long)ET * H1, TB), TB, 0, stream>>>(ei, E, ET, aS1, aD1, e1, (unsigned*)m1);
    edge_exppass<H1><<<cdiv_i((long long)ET * H1, TB), TB, 0, stream>>>(ei, E, ET, (const unsigned*)m1, e1, den1);
    edge_scatter<H1, C1><<<cdiv_i((long long)ET * (HC1 / 4), TB), TB, 0, stream>>>(ei, E, ET, h1, e1, den1, g1);

    bias_elu<256><<<cdiv_i((long long)N * HC1, TB), TB, 0, stream>>>(g1, b1, g1, (long long)N * HC1);

    // ================= layer 2 =================
    {   // h2 = g1 @ W2  (32 output cols -> one strip of 2 tiles per wave)
        int rowTiles = N / 16;
        long long waves = (long long)rowTiles;
        gemm_wmma_f32<256, 32, 2><<<cdiv_i(waves * 32, TB), TB, 0, stream>>>(g1, W2, h2, rowTiles);
    }
    att_scores<H2, C2><<<cdiv_i((long long)N * H2, TB), TB, 0, stream>>>(h2, atS2, atD2, aS2, aD2, N);

    fill_f32<<<cdiv_i((long long)N * 2, TB), TB, 0, stream>>>(m2, 0.f, (long long)N * 2); // m2+den2
    fill_f32<<<cdiv_i((long long)N * HC2, TB), TB, 0, stream>>>(out, 0.f, (long long)N * HC2);

    edge_maxpass<H2><<<cdiv_i((long long)ET * H2, TB), TB, 0, stream>>>(ei, E, ET, aS2, aD2, e2, (unsigned*)m2);
    edge_exppass<H2><<<cdiv_i((long long)ET * H2, TB), TB, 0, stream>>>(ei, E, ET, (const unsigned*)m2, e2, den2);
    edge_scatter<H2, C2><<<cdiv_i((long long)ET * (HC2 / 4), TB), TB, 0, stream>>>(ei, E, ET, h2, e2, den2, out);

    bias_elu<32><<<cdiv_i((long long)N * HC2, TB), TB, 0, stream>>>(out, b2, out, (long long)N * HC2);
}